// MultiHeadAttention_with_preference_23725399343405
// MI455X (gfx1250) — compile-verified
//
#include <hip/hip_runtime.h>
#include <math.h>

#if defined(__HIP_DEVICE_COMPILE__) && defined(__gfx1250__)
#if !__has_builtin(__builtin_amdgcn_wmma_f32_16x16x4_f32)
#error "device: missing __builtin_amdgcn_wmma_f32_16x16x4_f32"
#endif
#endif

typedef __attribute__((ext_vector_type(2))) float v2f;
typedef __attribute__((ext_vector_type(8))) float v8f;

// D = A(16x4 f32) * B(4x16 f32) + C(16x16 f32), full fp32 matrix pipe
#define WMMA_F32(a, b, c) \
  __builtin_amdgcn_wmma_f32_16x16x4_f32(false, (a), false, (b), (short)0, (c), false, false)

static constexpr int Bn = 2, Mn = 10, Nn = 1024, Dn = 128, Hn = 8, HDn = 16;

// ---------------------------------------------------------------------------
// Kernel 1: qkv = node_emb @ Wqkv + b   (2048x128)@(128x384)
// one 16x16 output tile per wave; 128 M-tiles x 24 N-tiles = 3072 waves
// ---------------------------------------------------------------------------
__global__ void __launch_bounds__(256) k_qkv(const float* __restrict__ node,
                                             const float* __restrict__ W,
                                             const float* __restrict__ bias,
                                             float* __restrict__ qkv) {
  const int lane = threadIdx.x & 31;
  const int wave = threadIdx.x >> 5;
  const int tile = blockIdx.x * 8 + wave;
  const int mt = tile & 127;   // row tile (B*N/16 = 128)
  const int nt = tile >> 7;    // col tile (3D/16 = 24)
  const int lo = lane & 15, hi = lane >> 4;
  const int row = mt * 16 + lo;       // A-operand row for this lane
  const int col = nt * 16 + lo;       // B/C-operand column for this lane
  const float* arow = node + (size_t)row * Dn;
  v8f c = {};
#pragma unroll
  for (int k0 = 0; k0 < Dn; k0 += 4) {
    v2f a = *(const v2f*)(arow + k0 + 2 * hi);                  // A[row][k0+2hi .. +1]
    const float* bp = W + (size_t)(k0 + 2 * hi) * (3 * Dn) + col;
    v2f b; b.x = bp[0]; b.y = bp[3 * Dn];                       // B[k][col], B[k+1][col]
    c = WMMA_F32(a, b, c);
  }
  const float bb = bias[col];
#pragma unroll
  for (int r = 0; r < 8; ++r) {
    const int orow = mt * 16 + r + 8 * hi;                      // C row = r + 8*hi
    qkv[(size_t)orow * (3 * Dn) + col] = c[r] + bb;
  }
}

// ---------------------------------------------------------------------------
// Kernel 2: x5 = prefer_emb @ Wp + b + broadcast(q|k)   (20480x128)@(128x128)
// 1280 M-tiles x 8 N-tiles = 10240 waves per projection
// ---------------------------------------------------------------------------
__global__ void __launch_bounds__(256) k_pref(const float* __restrict__ pref,
                                              const float* __restrict__ Wp,
                                              const float* __restrict__ bias,
                                              const float* __restrict__ qkv,
                                              int qoff, float* __restrict__ dst) {
  const int lane = threadIdx.x & 31;
  const int wave = threadIdx.x >> 5;
  const int tile = blockIdx.x * 8 + wave;
  const int nt = tile & 7;     // 8 col tiles
  const int mt = tile >> 3;    // 1280 row tiles
  const int lo = lane & 15, hi = lane >> 4;
  const int row = mt * 16 + lo;
  const int col = nt * 16 + lo;
  const float* arow = pref + (size_t)row * Dn;
  v8f c = {};
#pragma unroll
  for (int k0 = 0; k0 < Dn; k0 += 4) {
    v2f a = *(const v2f*)(arow + k0 + 2 * hi);
    const float* bp = Wp + (size_t)(k0 + 2 * hi) * Dn + col;
    v2f b; b.x = bp[0]; b.y = bp[Dn];
    c = WMMA_F32(a, b, c);
  }
  const float bb = bias[col];
#pragma unroll
  for (int r = 0; r < 8; ++r) {
    const int orow = mt * 16 + r + 8 * hi;        // flat (b*M + m)*N + n
    const int n = orow & (Nn - 1);
    const int bidx = orow / (Mn * Nn);
    const float qv = qkv[(size_t)(bidx * Nn + n) * (3 * Dn) + qoff + col];
    dst[(size_t)orow * Dn + col] = c[r] + bb + qv;
  }
}

// ---------------------------------------------------------------------------
// Kernel 3: fused scores -> softmax(m) -> sum(m) -> attn @ v
// one wave per (b, h, q-tile); loops 64 k-tiles. scores never hit memory.
// ---------------------------------------------------------------------------
__global__ void __launch_bounds__(32) k_attn(const float* __restrict__ q5,
                                             const float* __restrict__ k5,
                                             const float* __restrict__ qkv,
                                             float* __restrict__ outh) {
  __shared__ float attn_lds[16 * 16];
  const int lane = threadIdx.x & 31;
  const int lo = lane & 15, hi = lane >> 4;
  int id = blockIdx.x;
  const int qt = id & 63; id >>= 6;
  const int h = id & 7;  const int b = id >> 3;
  const int hd0 = h * HDn;
  const float* vbase = qkv + 2 * Dn;   // v columns start at 256

  v8f outc = {};
  for (int kt = 0; kt < 64; ++kt) {
    // ---- scores[m] = (q5 tile) @ (k5 tile)^T, K = head_dim = 16 ----
    v8f sc[Mn] = {};
#pragma unroll
    for (int m = 0; m < Mn; ++m) {
      const float* qrow = q5 + (size_t)((b * Mn + m) * Nn + qt * 16 + lo) * Dn + hd0;
      const float* krow = k5 + (size_t)((b * Mn + m) * Nn + kt * 16 + lo) * Dn + hd0;
#pragma unroll
      for (int d0 = 0; d0 < HDn; d0 += 4) {
        v2f a  = *(const v2f*)(qrow + d0 + 2 * hi);
        v2f bb = *(const v2f*)(krow + d0 + 2 * hi);   // B[K=d][N=k] = k5[k][d]
        sc[m] = WMMA_F32(a, bb, sc[m]);
      }
    }
    // ---- per-element softmax over m, then sum over m ----
#pragma unroll
    for (int e = 0; e < 8; ++e) {
      float x[Mn];
      float mx = -3.4e38f;
#pragma unroll
      for (int m = 0; m < Mn; ++m) { x[m] = sc[m][e] * 0.25f; mx = fmaxf(mx, x[m]); }
      float s = 0.f;
#pragma unroll
      for (int m = 0; m < Mn; ++m) { x[m] = __expf(x[m] - mx); s += x[m]; }
      const float inv = __builtin_amdgcn_rcpf(s);
      float acc = 0.f;
#pragma unroll
      for (int m = 0; m < Mn; ++m) acc += x[m] * inv;
      attn_lds[(e + 8 * hi) * 16 + lo] = acc;        // C/D layout -> row-major tile
    }
    __syncthreads();   // single wave: cheap; orders LDS write -> cross-lane read
    // ---- out += attn(16x16) @ v(16x16) ----
    const float* vrow = vbase + (size_t)(b * Nn + kt * 16) * (3 * Dn) + hd0;
#pragma unroll
    for (int d0 = 0; d0 < 16; d0 += 4) {
      v2f a = *(const v2f*)(&attn_lds[lo * 16 + d0 + 2 * hi]);    // A[q=lo][K=d0+2hi]
      const float* bp = vrow + (size_t)(d0 + 2 * hi) * (3 * Dn) + lo;
      v2f bb; bb.x = bp[0]; bb.y = bp[3 * Dn];
      outc = WMMA_F32(a, bb, outc);
    }
    __syncthreads();   // protect attn_lds before next k-tile overwrites
  }
#pragma unroll
  for (int r = 0; r < 8; ++r) {
    const int q = qt * 16 + r + 8 * hi;
    outh[(size_t)(b * Nn + q) * Dn + hd0 + lo] = outc[r];
  }
}

// ---------------------------------------------------------------------------
// Kernel 4: out = outh @ out_w + out_b   (2048x128)@(128x128)
// ---------------------------------------------------------------------------
__global__ void __launch_bounds__(256) k_out(const float* __restrict__ x,
                                             const float* __restrict__ W,
                                             const float* __restrict__ bias,
                                             float* __restrict__ y) {
  const int lane = threadIdx.x & 31;
  const int wave = threadIdx.x >> 5;
  const int tile = blockIdx.x * 8 + wave;
  const int nt = tile & 7;
  const int mt = tile >> 3;    // 128 row tiles
  const int lo = lane & 15, hi = lane >> 4;
  const int row = mt * 16 + lo;
  const int col = nt * 16 + lo;
  const float* arow = x + (size_t)row * Dn;
  v8f c = {};
#pragma unroll
  for (int k0 = 0; k0 < Dn; k0 += 4) {
    v2f a = *(const v2f*)(arow + k0 + 2 * hi);
    const float* bp = W + (size_t)(k0 + 2 * hi) * Dn + col;
    v2f b; b.x = bp[0]; b.y = bp[Dn];
    c = WMMA_F32(a, b, c);
  }
  const float bb = bias[col];
#pragma unroll
  for (int r = 0; r < 8; ++r) {
    const int orow = mt * 16 + r + 8 * hi;
    y[(size_t)orow * Dn + col] = c[r] + bb;
  }
}

// ---------------------------------------------------------------------------
extern "C" void kernel_launch(void* const* d_in, const int* in_sizes, int n_in,
                              void* d_out, int out_size, void* d_ws, size_t ws_size,
                              hipStream_t stream) {
  (void)in_sizes; (void)n_in; (void)out_size; (void)ws_size;
  const float* node = (const float*)d_in[0];
  const float* pref = (const float*)d_in[1];
  const float* Wqkv = (const float*)d_in[2];
  const float* bqkv = (const float*)d_in[3];
  const float* Wq   = (const float*)d_in[4];
  const float* bq   = (const float*)d_in[5];
  const float* Wk   = (const float*)d_in[6];
  const float* bk   = (const float*)d_in[7];
  const float* Wo   = (const float*)d_in[8];
  const float* bo   = (const float*)d_in[9];
  float* out = (float*)d_out;

  float* ws   = (float*)d_ws;
  float* qkv  = ws;                                   // 2*1024*384      = 786432 f
  float* q5   = qkv + (size_t)Bn * Nn * 3 * Dn;       // 2*10*1024*128   = 2621440 f
  float* k5   = q5  + (size_t)Bn * Mn * Nn * Dn;      // 2621440 f
  float* outh = k5  + (size_t)Bn * Mn * Nn * Dn;      // 2*1024*128      = 262144 f
                                                      // total ~25.2 MB (L2-resident)

  k_qkv <<<384, 256, 0, stream>>>(node, Wqkv, bqkv, qkv);
  k_pref<<<1280, 256, 0, stream>>>(pref, Wq, bq, qkv, 0,  q5);
  k_pref<<<1280, 256, 0, stream>>>(pref, Wk, bk, qkv, Dn, k5);
  k_attn<<<Bn * Hn * 64, 32, 0, stream>>>(q5, k5, qkv, outh);
  k_out <<<128, 256, 0, stream>>>(outh, Wo, bo, out);
}